// FastRotation_68942815035562
// MI455X (gfx1250) — compile-verified
//
#include <hip/hip_runtime.h>
#include <cmath>

#ifndef __has_builtin
#define __has_builtin(x) 0
#endif

#if __has_builtin(__builtin_amdgcn_global_load_async_to_lds_b32)
#define USE_ASYNC_LDS 1
#else
#define USE_ASYNC_LDS 0
#endif

#define KDIM 5
#define NPTS 125   // KDIM^3
#define FPB  2     // filters per block

typedef __attribute__((address_space(1))) int g_int;  // global
typedef __attribute__((address_space(3))) int l_int;  // LDS

__global__ __launch_bounds__(128 * FPB) void FastRotation_68942815035562_kernel(
    const float* __restrict__ vol,      // (N,5,5,5)
    const float* __restrict__ theta_v,  // (N,3)
    const float* __restrict__ theta,    // (N,)
    float* __restrict__ out,            // (N,5,5,5)
    int N)
{
  const int tid  = threadIdx.x;
  const int half = tid >> 7;          // which filter within the block
  const int t    = tid & 127;         // lane within the filter's 128 threads
  const int n    = blockIdx.x * FPB + half;
  const bool live = (n < N);

  __shared__ float sv[FPB][128];      // volumes (125 used, padded to 128)
  __shared__ float sR[FPB][12];       // rotation matrices (9 used, padded)

  const float* __restrict__ voln = vol + (size_t)n * NPTS;

  // Stage each 125-float volume into LDS via the CDNA5 async global->LDS path
  // (each voxel is reused ~8x by the trilinear taps).
  if (live && t < NPTS) {
#if USE_ASYNC_LDS
    __builtin_amdgcn_global_load_async_to_lds_b32(
        (g_int*)(voln + t),
        (l_int*)&sv[half][t],
        /*imm offset*/ 0, /*cpol*/ 0);
#else
    sv[half][t] = __builtin_nontemporal_load(voln + t);
#endif
  }

  // One Rodrigues matrix per filter; computed once (one fast sincos each).
  if (live && t == 0) {
    float vx = theta_v[3 * n + 0];
    float vy = theta_v[3 * n + 1];
    float vz = theta_v[3 * n + 2];
    const float nrm = sqrtf(vx * vx + vy * vy + vz * vz);
    const float inv = 1.0f / fmaxf(nrm, 1e-12f);
    vx *= inv; vy *= inv; vz *= inv;
    const float th = theta[n];
    const float s  = __sinf(th);          // theta ~ N(0,1): HW trans path is exact enough
    const float c  = 1.0f - __cosf(th);
    // m = skew(v); mm = m*m in closed form: diag = -(sum of other squares),
    // off-diag = v_a*v_b  (same structure as the reference einsum).
    const float mm00 = -(vz * vz + vy * vy);
    const float mm11 = -(vz * vz + vx * vx);
    const float mm22 = -(vy * vy + vx * vx);
    const float mxy = vx * vy, mxz = vx * vz, myz = vy * vz;
    sR[half][0] = 1.0f + c * mm00;       // R00
    sR[half][1] = -s * vz + c * mxy;     // R01
    sR[half][2] =  s * vy + c * mxz;     // R02
    sR[half][3] =  s * vz + c * mxy;     // R10
    sR[half][4] = 1.0f + c * mm11;       // R11
    sR[half][5] = -s * vx + c * myz;     // R12
    sR[half][6] = -s * vy + c * mxz;     // R20
    sR[half][7] =  s * vx + c * myz;     // R21
    sR[half][8] = 1.0f + c * mm22;       // R22
  }

#if USE_ASYNC_LDS
#if __has_builtin(__builtin_amdgcn_s_wait_asynccnt)
  __builtin_amdgcn_s_wait_asynccnt(0);
#else
  asm volatile("s_wait_asynccnt 0" ::: "memory");
#endif
#endif
  __syncthreads();

  if (live && t < NPTS) {
    // p -> (i,j,l) in meshgrid 'ij' order; base coord c = linspace(-1,1,5).
    const int i = t / 25;
    const int r = t - i * 25;
    const int j = r / 5;
    const int l = r - j * 5;
    const float bi = -1.0f + 0.5f * (float)i;
    const float bj = -1.0f + 0.5f * (float)j;
    const float bl = -1.0f + 0.5f * (float)l;

    // grid[k] = sum_j base[j] * R[j][k]
    const float gx = bi * sR[half][0] + bj * sR[half][3] + bl * sR[half][6];
    const float gy = bi * sR[half][1] + bj * sR[half][4] + bl * sR[half][7];
    const float gz = bi * sR[half][2] + bj * sR[half][5] + bl * sR[half][8];

    // (g+1)*0.5*(dim-1), dim-1 = 4 -> exact power-of-two scaling
    const float x = (gx + 1.0f) * 2.0f;
    const float y = (gy + 1.0f) * 2.0f;
    const float z = (gz + 1.0f) * 2.0f;
    const float x0f = floorf(x), y0f = floorf(y), z0f = floorf(z);
    const float wx = x - x0f, wy = y - y0f, wz = z - z0f;
    const int x0 = (int)x0f, y0 = (int)y0f, z0 = (int)z0f;

    float acc = 0.0f;
#pragma unroll
    for (int dz = 0; dz < 2; ++dz) {
      const int zi = z0 + dz;
      const float fz = dz ? wz : 1.0f - wz;
      const bool zb = (zi >= 0) & (zi < KDIM);
      const int zc = zi < 0 ? 0 : (zi > KDIM - 1 ? KDIM - 1 : zi);
#pragma unroll
      for (int dy = 0; dy < 2; ++dy) {
        const int yi = y0 + dy;
        const float fy = dy ? wy : 1.0f - wy;
        const bool yb = (yi >= 0) & (yi < KDIM);
        const int yc = yi < 0 ? 0 : (yi > KDIM - 1 ? KDIM - 1 : yi);
#pragma unroll
        for (int dx = 0; dx < 2; ++dx) {
          const int xi = x0 + dx;
          const float fx = dx ? wx : 1.0f - wx;
          const bool xb = (xi >= 0) & (xi < KDIM);
          const int xc = xi < 0 ? 0 : (xi > KDIM - 1 ? KDIM - 1 : xi);
          const float v = sv[half][zc * 25 + yc * 5 + xc];
          const float w = fz * fy * fx;
          acc += w * ((zb & yb & xb) ? v : 0.0f);
        }
      }
    }
    // Streaming output (131 MB): NT store keeps it from thrashing the 192MB L2.
    __builtin_nontemporal_store(acc, out + (size_t)n * NPTS + t);
  }
}

extern "C" void kernel_launch(void* const* d_in, const int* in_sizes, int n_in,
                              void* d_out, int out_size, void* d_ws, size_t ws_size,
                              hipStream_t stream) {
  (void)n_in; (void)out_size; (void)d_ws; (void)ws_size;
  const float* vol     = (const float*)d_in[0];
  const float* theta_v = (const float*)d_in[1];
  const float* theta   = (const float*)d_in[2];
  float* out           = (float*)d_out;
  const int N = in_sizes[2];  // one theta per filter
  const unsigned blocks = (unsigned)((N + FPB - 1) / FPB);
  FastRotation_68942815035562_kernel<<<dim3(blocks), dim3(128 * FPB), 0, stream>>>(
      vol, theta_v, theta, out, N);
}